// AttenedDecoderRnn_82403242541518
// MI455X (gfx1250) — compile-verified
//
#include <hip/hip_runtime.h>
#include <math.h>

#define HD 1024
#define VD 32000
#define SD 4096

typedef __attribute__((ext_vector_type(2))) float v2f;
typedef __attribute__((ext_vector_type(8))) float v8f;

__device__ __forceinline__ float wave_sum(float v) {
#pragma unroll
  for (int o = 16; o > 0; o >>= 1) v += __shfl_xor(v, o, 32);
  return v;
}
__device__ __forceinline__ float wave_max(float v) {
#pragma unroll
  for (int o = 16; o > 0; o >>= 1) v = fmaxf(v, __shfl_xor(v, o, 32));
  return v;
}

// one wave computes dot(w[0..n), x[0..n)) with float4 loads
__device__ __forceinline__ float wave_dot(const float* __restrict__ w,
                                          const float* __restrict__ x,
                                          int n, int lane) {
  float acc = 0.f;
  for (int j = lane * 4; j < n; j += 128) {
    const float4 a = *(const float4*)(w + j);
    const float4 b = *(const float4*)(x + j);
    acc = fmaf(a.x, b.x, fmaf(a.y, b.y, fmaf(a.z, b.z, fmaf(a.w, b.w, acc))));
  }
  return wave_sum(acc);
}

// ---------- Kernel A: fused GRU cell -> h1 (one wave per output element) ----------
__global__ __launch_bounds__(256) void gru_kernel(
    const int* __restrict__ word, const float* __restrict__ h0,
    const float* __restrict__ emb, const float* __restrict__ lastctx,
    const float* __restrict__ W_ih, const float* __restrict__ W_hh,
    const float* __restrict__ b_ih, const float* __restrict__ b_hh,
    float* __restrict__ ws_h1, float* __restrict__ out_h1) {
  __shared__ float xs[2 * HD];
  __shared__ float hs[HD];
  const int tid = threadIdx.x;
  const int w = word[0];
  for (int j = tid; j < HD; j += 256) {
    xs[j] = emb[(size_t)w * HD + j];
    xs[HD + j] = lastctx[j];
    hs[j] = h0[j];
  }
  __syncthreads();
  const int lane = tid & 31;
  const int i = blockIdx.x * 8 + (tid >> 5);  // 128 blocks * 8 waves = 1024
  const float gir = wave_dot(W_ih + (size_t)i * 2 * HD, xs, 2 * HD, lane) + b_ih[i];
  const float giz = wave_dot(W_ih + (size_t)(HD + i) * 2 * HD, xs, 2 * HD, lane) + b_ih[HD + i];
  const float gin = wave_dot(W_ih + (size_t)(2 * HD + i) * 2 * HD, xs, 2 * HD, lane) + b_ih[2 * HD + i];
  const float ghr = wave_dot(W_hh + (size_t)i * HD, hs, HD, lane) + b_hh[i];
  const float ghz = wave_dot(W_hh + (size_t)(HD + i) * HD, hs, HD, lane) + b_hh[HD + i];
  const float ghn = wave_dot(W_hh + (size_t)(2 * HD + i) * HD, hs, HD, lane) + b_hh[2 * HD + i];
  if (lane == 0) {
    const float r = 1.f / (1.f + expf(-(gir + ghr)));
    const float z = 1.f / (1.f + expf(-(giz + ghz)));
    const float n = tanhf(gin + r * ghn);
    const float h1 = (1.f - z) * n + z * hs[i];
    ws_h1[i] = h1;
    out_h1[i] = h1;
  }
}

// ---------- Kernel B: v = W_att^T @ h1 ----------
__global__ __launch_bounds__(128) void attv_kernel(const float* __restrict__ W_att,
                                                   const float* __restrict__ ws_h1,
                                                   float* __restrict__ ws_v) {
  __shared__ float hsm[HD];
  for (int j = threadIdx.x; j < HD; j += 128) hsm[j] = ws_h1[j];
  __syncthreads();
  const int col = blockIdx.x * 128 + threadIdx.x;  // 8 blocks
  float acc = 0.f;
  for (int i = 0; i < HD; i += 4) {
    acc = fmaf(W_att[(size_t)i * HD + col], hsm[i], acc);
    acc = fmaf(W_att[(size_t)(i + 1) * HD + col], hsm[i + 1], acc);
    acc = fmaf(W_att[(size_t)(i + 2) * HD + col], hsm[i + 2], acc);
    acc = fmaf(W_att[(size_t)(i + 3) * HD + col], hsm[i + 3], acc);
  }
  ws_v[col] = acc;
}

// ---------- Kernel C: scores[s] = dot(enc[s], v) ----------
__global__ __launch_bounds__(256) void scores_kernel(const float* __restrict__ enc,
                                                     const float* __restrict__ ws_v,
                                                     float* __restrict__ ws_scores) {
  __shared__ float vs[HD];
  for (int j = threadIdx.x; j < HD; j += 256) vs[j] = ws_v[j];
  __syncthreads();
  const int lane = threadIdx.x & 31;
  const int gw = blockIdx.x * 8 + (threadIdx.x >> 5);  // 64 blocks -> 512 waves
  for (int t = 0; t < 8; ++t) {
    const int s = gw * 8 + t;
    const float d = wave_dot(enc + (size_t)s * HD, vs, HD, lane);
    if (lane == 0) ws_scores[s] = d;
  }
}

// ---------- Kernel D: softmax over 4096 scores ----------
__global__ __launch_bounds__(1024) void softmax_kernel(const float* __restrict__ ws_scores,
                                                       float* __restrict__ ws_attn,
                                                       float* __restrict__ out_attn) {
  __shared__ float red[32];
  __shared__ float Mv, Zv;
  const int tid = threadIdx.x;
  const float x0 = ws_scores[tid], x1 = ws_scores[1024 + tid];
  const float x2 = ws_scores[2048 + tid], x3 = ws_scores[3072 + tid];
  float m = fmaxf(fmaxf(x0, x1), fmaxf(x2, x3));
  m = wave_max(m);
  if ((tid & 31) == 0) red[tid >> 5] = m;
  __syncthreads();
  if (tid < 32) {
    float t = red[tid];
    t = wave_max(t);
    if (tid == 0) Mv = t;
  }
  __syncthreads();
  const float M = Mv;
  const float e0 = expf(x0 - M), e1 = expf(x1 - M), e2 = expf(x2 - M), e3 = expf(x3 - M);
  float s = e0 + e1 + e2 + e3;
  s = wave_sum(s);
  if ((tid & 31) == 0) red[tid >> 5] = s;
  __syncthreads();
  if (tid < 32) {
    float t = red[tid];
    t = wave_sum(t);
    if (tid == 0) Zv = t;
  }
  __syncthreads();
  const float inv = 1.f / Zv;
  ws_attn[tid] = e0 * inv;            out_attn[tid] = e0 * inv;
  ws_attn[1024 + tid] = e1 * inv;     out_attn[1024 + tid] = e1 * inv;
  ws_attn[2048 + tid] = e2 * inv;     out_attn[2048 + tid] = e2 * inv;
  ws_attn[3072 + tid] = e3 * inv;     out_attn[3072 + tid] = e3 * inv;
}

// ---------- Kernel E: context partials (16 S-chunks x 1024 cols) ----------
__global__ __launch_bounds__(256) void context_kernel(const float* __restrict__ enc,
                                                      const float* __restrict__ ws_attn,
                                                      float* __restrict__ ws_cpart) {
  const int cchunk = blockIdx.x & 3;
  const int schunk = blockIdx.x >> 2;  // 64 blocks
  const int col = cchunk * 256 + threadIdx.x;
  const int s0 = schunk * 256;
  float acc = 0.f;
  for (int s = s0; s < s0 + 256; ++s)
    acc = fmaf(ws_attn[s], enc[(size_t)s * HD + col], acc);
  ws_cpart[schunk * HD + col] = acc;
}

__global__ __launch_bounds__(256) void ctxreduce_kernel(const float* __restrict__ ws_cpart,
                                                        float* __restrict__ ws_ctx,
                                                        float* __restrict__ out_ctx) {
  const int col = blockIdx.x * 256 + threadIdx.x;  // 4 blocks
  float s = 0.f;
#pragma unroll
  for (int c = 0; c < 16; ++c) s += ws_cpart[c * HD + col];
  ws_ctx[col] = s;
  out_ctx[col] = s;
}

// ---------- Kernel F: logits = W_fc @ [h1;context] + b_fc via V_WMMA_F32_16X16X4_F32 ----------
// Each wave owns 16 rows of W_fc. A tile: 16x4 fp32 per ISA layout
// (lanes 0-15 hold K=k0,k0+1; lanes 16-31 hold K=k0+2,k0+3). B replicates the
// y vector across all N columns, so D column 0 = the 16 dot products.
__global__ __launch_bounds__(256) void logits_kernel(const float* __restrict__ W_fc,
                                                     const float* __restrict__ b_fc,
                                                     const float* __restrict__ ws_y,
                                                     float* __restrict__ out_logits) {
  __shared__ float ys[2 * HD];
  for (int j = threadIdx.x; j < 2 * HD; j += 256) ys[j] = ws_y[j];
  __syncthreads();
  const int lane = threadIdx.x & 31;
  const int tile = blockIdx.x * 8 + (threadIdx.x >> 5);  // 250 blocks * 8 waves = 2000 tiles
  const int base = tile * 16;
  const int koff = (lane >> 4) << 1;  // 0 for lanes 0-15, 2 for lanes 16-31
  const float* wrow = W_fc + (size_t)(base + (lane & 15)) * (2 * HD);
  v8f c = {};
#pragma unroll 4
  for (int k0 = 0; k0 < 2 * HD; k0 += 4) {
    const float2 av = *(const float2*)(wrow + k0 + koff);
    v2f a, b;
    a.x = av.x;
    a.y = av.y;
    b.x = ys[k0 + koff];
    b.y = ys[k0 + koff + 1];
    c = __builtin_amdgcn_wmma_f32_16x16x4_f32(false, a, false, b, (short)0, c,
                                              false, false);
  }
  // D layout: VGPR r holds (M=r, N=lane) on lanes 0-15 and (M=8+r) on lanes 16-31.
  if (lane == 0) {
#pragma unroll
    for (int r = 0; r < 8; ++r) out_logits[base + r] = c[r] + b_fc[base + r];
  } else if (lane == 16) {
#pragma unroll
    for (int r = 0; r < 8; ++r) out_logits[base + 8 + r] = c[r] + b_fc[base + 8 + r];
  }
}

// ---------- log-softmax over 32000 logits (3 stages, in place on d_out) ----------
__global__ __launch_bounds__(256) void maxpart_kernel(const float* __restrict__ logits,
                                                      float* __restrict__ ws_pmax) {
  __shared__ float red[8];
  const int b = blockIdx.x;  // 32 blocks x 1000 elems
  float m = -INFINITY;
  for (int idx = b * 1000 + threadIdx.x; idx < (b + 1) * 1000; idx += 256)
    m = fmaxf(m, logits[idx]);
  m = wave_max(m);
  if ((threadIdx.x & 31) == 0) red[threadIdx.x >> 5] = m;
  __syncthreads();
  if (threadIdx.x == 0) {
    float t = red[0];
    for (int i = 1; i < 8; ++i) t = fmaxf(t, red[i]);
    ws_pmax[b] = t;
  }
}

__global__ __launch_bounds__(256) void sumexp_kernel(const float* __restrict__ logits,
                                                     const float* __restrict__ ws_pmax,
                                                     float* __restrict__ ws_psum) {
  __shared__ float red[8];
  float M = ws_pmax[0];
  for (int i = 1; i < 32; ++i) M = fmaxf(M, ws_pmax[i]);
  const int b = blockIdx.x;
  float s = 0.f;
  for (int idx = b * 1000 + threadIdx.x; idx < (b + 1) * 1000; idx += 256)
    s += expf(logits[idx] - M);
  s = wave_sum(s);
  if ((threadIdx.x & 31) == 0) red[threadIdx.x >> 5] = s;
  __syncthreads();
  if (threadIdx.x == 0) {
    float t = 0.f;
    for (int i = 0; i < 8; ++i) t += red[i];
    ws_psum[b] = t;
  }
}

__global__ __launch_bounds__(256) void finalize_kernel(float* __restrict__ logits,
                                                       const float* __restrict__ ws_pmax,
                                                       const float* __restrict__ ws_psum) {
  float M = ws_pmax[0];
  for (int i = 1; i < 32; ++i) M = fmaxf(M, ws_pmax[i]);
  float Z = 0.f;
  for (int i = 0; i < 32; ++i) Z += ws_psum[i];
  const float logZ = logf(Z);
  const int idx = blockIdx.x * 256 + threadIdx.x;  // 125 blocks
  if (idx < VD) logits[idx] = logits[idx] - M - logZ;
}

extern "C" void kernel_launch(void* const* d_in, const int* in_sizes, int n_in,
                              void* d_out, int out_size, void* d_ws, size_t ws_size,
                              hipStream_t stream) {
  const int*   word    = (const int*)d_in[0];
  const float* h0      = (const float*)d_in[1];   // decoder_last_hidden (1,1,H)
  const float* enc     = (const float*)d_in[2];   // encoder_outputs (S,1,H)
  const float* lastctx = (const float*)d_in[3];   // (1,H)
  const float* emb     = (const float*)d_in[4];   // (V,H)
  const float* W_ih    = (const float*)d_in[5];
  const float* W_hh    = (const float*)d_in[6];
  const float* b_ih    = (const float*)d_in[7];
  const float* b_hh    = (const float*)d_in[8];
  const float* W_att   = (const float*)d_in[9];
  // d_in[10] = b_att: algebraically cancels (constant shift before softmax)
  const float* W_fc    = (const float*)d_in[11];
  const float* b_fc    = (const float*)d_in[12];

  float* out = (float*)d_out;
  float* ws  = (float*)d_ws;

  // workspace layout (floats); y = ws[0..2047] = [h1 | context]
  float* ws_h1     = ws;            // 1024
  float* ws_ctx    = ws + 1024;     // 1024
  float* ws_scores = ws + 2048;     // 4096
  float* ws_attn   = ws + 6144;     // 4096
  float* ws_v      = ws + 10240;    // 1024
  float* ws_pmax   = ws + 11264;    // 32
  float* ws_psum   = ws + 11296;    // 32
  float* ws_cpart  = ws + 11392;    // 16*1024

  // output layout: log_softmax(32000) | context(1024) | h1(1024) | attn(4096)
  float* out_logits = out;
  float* out_ctx    = out + 32000;
  float* out_h1     = out + 33024;
  float* out_attn   = out + 34048;

  gru_kernel<<<128, 256, 0, stream>>>(word, h0, emb, lastctx, W_ih, W_hh, b_ih, b_hh, ws_h1, out_h1);
  attv_kernel<<<8, 128, 0, stream>>>(W_att, ws_h1, ws_v);
  scores_kernel<<<64, 256, 0, stream>>>(enc, ws_v, ws_scores);
  softmax_kernel<<<1, 1024, 0, stream>>>(ws_scores, ws_attn, out_attn);
  context_kernel<<<64, 256, 0, stream>>>(enc, ws_attn, ws_cpart);
  ctxreduce_kernel<<<4, 256, 0, stream>>>(ws_cpart, ws_ctx, out_ctx);
  logits_kernel<<<250, 256, 0, stream>>>(W_fc, b_fc, ws_h1, out_logits);
  maxpart_kernel<<<32, 256, 0, stream>>>(out_logits, ws_pmax);
  sumexp_kernel<<<32, 256, 0, stream>>>(out_logits, ws_pmax, ws_psum);
  finalize_kernel<<<125, 256, 0, stream>>>(out_logits, ws_pmax, ws_psum);
}